// SelfAttentionLayer_53996328845312
// MI455X (gfx1250) — compile-verified
//
#include <hip/hip_runtime.h>
#include <hip/hip_bf16.h>
#include <math.h>
#include <stdint.h>

#define N_NODES 50000
#define N_EDGES 800000
#define D_MODEL 64
#define N_HEADS 4

typedef __attribute__((ext_vector_type(2))) float v2f;
typedef __attribute__((ext_vector_type(8))) float v8f;

#define STRIDE_A 68   // A-tile LDS stride: banks (4m+k), conflict-free for both half-waves

// ---------- helpers ----------

__device__ inline float wred_sum(float v) {
    v += __shfl_xor(v, 16, 32);
    v += __shfl_xor(v, 8, 32);
    v += __shfl_xor(v, 4, 32);
    v += __shfl_xor(v, 2, 32);
    v += __shfl_xor(v, 1, 32);
    return v;
}

// order-preserving float <-> int map for atomicMax on floats
__device__ inline int fflip(float f) {
    int i = __float_as_int(f);
    return (i < 0) ? (i ^ 0x7fffffff) : i;
}
__device__ inline float funflip(int i) {
    return __int_as_float((i < 0) ? (i ^ 0x7fffffff) : i);
}

// async 16B copy global -> LDS (tracked by ASYNCcnt)
__device__ inline void async_copy_b128(float* lds_dst, const float* g_src) {
    uint32_t laddr = (uint32_t)(uintptr_t)lds_dst;
    uint64_t gaddr = (uint64_t)(uintptr_t)g_src;
    asm volatile("global_load_async_to_lds_b128 %0, %1, off"
                 :: "v"(laddr), "v"(gaddr) : "memory");
}
__device__ inline void async_wait0() {
    asm volatile("s_wait_asynccnt 0x0" ::: "memory");
}

__device__ inline float gelu_exact(float v) {
    return 0.5f * v * (1.0f + erff(v * 0.70710678118654752f));
}

// ---------- kernels ----------

__global__ void k_init_stats(int* gmax, float* gsum) {
    int t = threadIdx.x;
    if (t < N_HEADS) {
        gmax[t] = fflip(-INFINITY);
        gsum[t] = 0.0f;
    }
}

// Fused LayerNorm + GEMM: Out[N, ncols] = LN(X) @ W (+bias)(+gelu)
// 64 nodes/block, 256 threads (8 waves). Weights staged to LDS in two
// column-halves via async global->LDS; A-fragments register-cached per wave.
__global__ void k_ln_gemm(const float* __restrict__ X,
                          const float* __restrict__ g,
                          const float* __restrict__ bta,
                          const float* __restrict__ W,
                          const float* __restrict__ bias,
                          float* __restrict__ Out,
                          int ncols, int act) {
    extern __shared__ float smem[];
    float* ldsA = smem;                       // 64 x STRIDE_A
    float* ldsB = smem + 64 * STRIDE_A;       // 64 x (hc+16)
    const int hc  = ncols >> 1;               // 96 or 128 columns per pass
    const int stB = hc + 16;                  // 112 / 144: st ≡ 16 (mod 32)
    const int tid  = threadIdx.x;
    const int lane = tid & 31;
    const int wid  = tid >> 5;
    const int n0   = blockIdx.x * 64;

    // --- LayerNorm: each wave normalizes 8 rows into ldsA ---
    for (int i = 0; i < 8; ++i) {
        int r = wid * 8 + i;
        int n = n0 + r;
        float a0 = 0.f, a1 = 0.f;
        if (n < N_NODES) {
            a0 = X[n * 64 + lane];
            a1 = X[n * 64 + 32 + lane];
        }
        float mean = wred_sum(a0 + a1) * (1.0f / 64.0f);
        float d0 = a0 - mean, d1 = a1 - mean;
        float var = wred_sum(d0 * d0 + d1 * d1) * (1.0f / 64.0f);
        float rs = rsqrtf(var + 1e-5f);
        ldsA[r * STRIDE_A + lane]      = d0 * rs * g[lane]      + bta[lane];
        ldsA[r * STRIDE_A + 32 + lane] = d1 * rs * g[32 + lane] + bta[32 + lane];
    }
    __syncthreads();

    // --- register-cache A fragments for this wave's row tile ---
    const int rt    = wid & 3;                // 4 row tiles, 2 waves each
    const int nbase = n0 + rt * 16;
    const bool rv   = (nbase < N_NODES);      // wave-uniform (N % 16 == 0)
    const int m  = lane & 15;
    const int kh = (lane & 16) ? 2 : 0;
    v2f af[16];
#pragma unroll
    for (int k = 0; k < 16; ++k) {
        af[k].x = ldsA[(rt * 16 + m) * STRIDE_A + 4 * k + kh];
        af[k].y = ldsA[(rt * 16 + m) * STRIDE_A + 4 * k + kh + 1];
    }

    const int nf4 = 64 * hc / 4;
    const int cw  = hc / 4;                   // float4s per row
    for (int half = 0; half < 2; ++half) {
        __syncthreads();                      // previous half's B fully consumed
        for (int f = tid; f < nf4; f += 256) {
            int row = f / cw, c4 = f % cw;
            async_copy_b128(ldsB + row * stB + c4 * 4,
                            W + row * ncols + half * hc + c4 * 4);
        }
        async_wait0();
        __syncthreads();

        if (rv) {
            const int ctiles = hc >> 4;       // 6 or 8
            for (int ct = (wid >> 2); ct < ctiles; ct += 2) {
                v8f acc = {};
#pragma unroll
                for (int k = 0; k < 16; ++k) {
                    v2f b;
                    b.x = ldsB[(4 * k + kh) * stB + ct * 16 + m];
                    b.y = ldsB[(4 * k + kh + 1) * stB + ct * 16 + m];
                    acc = __builtin_amdgcn_wmma_f32_16x16x4_f32(
                        false, af[k], false, b, (short)0, acc, false, false);
                }
                int col = half * hc + ct * 16 + m;
                int rb  = (lane & 16) ? 8 : 0;
                float bv = bias ? bias[col] : 0.0f;
#pragma unroll
                for (int j = 0; j < 8; ++j) {
                    float v = acc[j] + bv;
                    if (act) v = gelu_exact(v);
                    Out[(nbase + rb + j) * ncols + col] = v;
                }
            }
        }
    }
}

// Per-(edge,head) attention score + global max (stable softmax over ALL edges).
__global__ void k_scores(const float* __restrict__ qkv,
                         const int* __restrict__ ei,
                         float* __restrict__ scores,
                         int* __restrict__ gmax) {
    __shared__ int lmax[N_HEADS];
    const int tid = threadIdx.x;
    if (tid < N_HEADS) lmax[tid] = fflip(-INFINITY);
    __syncthreads();

    int idx = blockIdx.x * 256 + tid;
    if (idx < N_EDGES * N_HEADS) {
        int e = idx >> 2, h = idx & 3;
        int src = ei[e];
        int dst = ei[N_EDGES + e];
        const float4* qp = (const float4*)(qkv + src * 192 + h * 16);
        const float4* kp = (const float4*)(qkv + dst * 192 + 64 + h * 16);
        float dot = 0.f;
#pragma unroll
        for (int i = 0; i < 4; ++i) {
            float4 a = qp[i], b = kp[i];
            dot += a.x * b.x + a.y * b.y + a.z * b.z + a.w * b.w;
        }
        float s = dot * 0.25f;   // 1/sqrt(16)
        scores[idx] = s;
        atomicMax(&lmax[h], fflip(s));
    }
    __syncthreads();
    if (tid < N_HEADS) atomicMax(&gmax[tid], lmax[tid]);
}

// exp(s - max): scatter into per-node-per-head S[dst][h] and global denom.
__global__ void k_expsum(const float* __restrict__ scores,
                         const int* __restrict__ ei,
                         const int* __restrict__ gmax,
                         float* __restrict__ S,
                         float* __restrict__ gsum) {
    __shared__ float lsum[N_HEADS];
    const int tid = threadIdx.x;
    if (tid < N_HEADS) lsum[tid] = 0.0f;
    __syncthreads();

    int idx = blockIdx.x * 256 + tid;
    if (idx < N_EDGES * N_HEADS) {
        int e = idx >> 2, h = idx & 3;
        float w = expf(scores[idx] - funflip(gmax[h]));
        int dst = ei[N_EDGES + e];
        atomicAdd(&S[dst * N_HEADS + h], w);
        atomicAdd(&lsum[h], w);
    }
    __syncthreads();
    if (tid < N_HEADS) atomicAdd(&gsum[tid], lsum[tid]);
}

// agg[n] = v[n] * S[n][h]/sum[h]; x1 = x + agg @ w_out + b_out  (WMMA GEMM)
__global__ void k_attn_out(const float* __restrict__ qkv,
                           const float* __restrict__ S,
                           const float* __restrict__ gsum,
                           const float* __restrict__ Wout,
                           const float* __restrict__ bout,
                           const float* __restrict__ X,
                           float* __restrict__ x1) {
    extern __shared__ float smem[];
    float* ldsA = smem;                    // 64 x STRIDE_A
    float* ldsB = smem + 64 * STRIDE_A;    // 64 x 80 (st ≡ 16 mod 32)
    const int stB = 80;
    const int tid  = threadIdx.x;
    const int lane = tid & 31;
    const int wid  = tid >> 5;
    const int n0   = blockIdx.x * 64;

    float inv[N_HEADS];
#pragma unroll
    for (int h = 0; h < N_HEADS; ++h) inv[h] = 1.0f / gsum[h];

    // stage w_out (64x64) into LDS asynchronously
    for (int f = tid; f < 64 * 16; f += 256) {
        int row = f >> 4, c4 = f & 15;
        async_copy_b128(ldsB + row * stB + c4 * 4, Wout + row * 64 + c4 * 4);
    }
    // build agg tile in ldsA
    for (int i = tid; i < 64 * 64; i += 256) {
        int r = i >> 6, c = i & 63;
        int n = n0 + r;
        float v = 0.f;
        if (n < N_NODES) {
            int h = c >> 4;
            v = qkv[n * 192 + 128 + c] * (S[n * N_HEADS + h] * inv[h]);
        }
        ldsA[r * STRIDE_A + c] = v;
    }
    async_wait0();
    __syncthreads();

    const int rt    = wid & 3;
    const int nbase = n0 + rt * 16;
    const int m  = lane & 15;
    const int kh = (lane & 16) ? 2 : 0;
    if (nbase < N_NODES) {                 // wave-uniform
        v2f af[16];
#pragma unroll
        for (int k = 0; k < 16; ++k) {
            af[k].x = ldsA[(rt * 16 + m) * STRIDE_A + 4 * k + kh];
            af[k].y = ldsA[(rt * 16 + m) * STRIDE_A + 4 * k + kh + 1];
        }
        for (int ct = (wid >> 2); ct < 4; ct += 2) {
            v8f acc = {};
#pragma unroll
            for (int k = 0; k < 16; ++k) {
                v2f b;
                b.x = ldsB[(4 * k + kh) * stB + ct * 16 + m];
                b.y = ldsB[(4 * k + kh + 1) * stB + ct * 16 + m];
                acc = __builtin_amdgcn_wmma_f32_16x16x4_f32(
                    false, af[k], false, b, (short)0, acc, false, false);
            }
            int col = ct * 16 + m;
            int rb  = (lane & 16) ? 8 : 0;
            float bv = bout[col];
#pragma unroll
            for (int j = 0; j < 8; ++j) {
                int row = nbase + rb + j;
                x1[row * 64 + col] = acc[j] + bv + X[row * 64 + col];
            }
        }
    }
}

// out = x1 + h @ w_ffn2 + b_ffn2 (K=256 WMMA GEMM), 32 nodes per block.
// Activation tile staged to LDS via async copy; W2 streamed from L2.
__global__ void k_ffn2(const float* __restrict__ hbuf,
                       const float* __restrict__ W2,
                       const float* __restrict__ b2,
                       const float* __restrict__ x1,
                       float* __restrict__ out) {
    extern __shared__ float smem[];
    float* ldsA = smem;                    // 32 x 260 (banks 4m+k, conflict-free)
    const int stA = 260;
    const int tid  = threadIdx.x;
    const int lane = tid & 31;
    const int wid  = tid >> 5;
    const int n0   = blockIdx.x * 32;

    for (int f = tid; f < 32 * 64; f += 256) {
        int row = f >> 6, c4 = f & 63;
        int n = n0 + row;
        if (n >= N_NODES) n = N_NODES - 1;   // rows beyond N only feed skipped tiles
        async_copy_b128(ldsA + row * stA + c4 * 4, hbuf + n * 256 + c4 * 4);
    }
    async_wait0();
    __syncthreads();

    // 2 row-tiles x 4 col-tiles = 8 tiles, exactly one per wave
    const int rt = wid >> 2, ct = wid & 3;
    const int nbase = n0 + rt * 16;
    const int m  = lane & 15;
    const int kh = (lane & 16) ? 2 : 0;
    if (nbase < N_NODES) {                 // wave-uniform
        v8f acc = {};
        const float* A = ldsA + rt * 16 * stA;
        const float* B = W2 + ct * 16;
#pragma unroll 4
        for (int k0 = 0; k0 < 256; k0 += 4) {
            v2f a, b;
            a.x = A[m * stA + k0 + kh];
            a.y = A[m * stA + k0 + kh + 1];
            b.x = B[(k0 + kh) * 64 + m];
            b.y = B[(k0 + kh + 1) * 64 + m];
            acc = __builtin_amdgcn_wmma_f32_16x16x4_f32(
                false, a, false, b, (short)0, acc, false, false);
        }
        int col = ct * 16 + m;
        int rb  = (lane & 16) ? 8 : 0;
        float bv = b2[col];
#pragma unroll
        for (int j = 0; j < 8; ++j) {
            int row = nbase + rb + j;
            out[row * 64 + col] = acc[j] + bv + x1[row * 64 + col];
        }
    }
}

// ---------- launch ----------

extern "C" void kernel_launch(void* const* d_in, const int* in_sizes, int n_in,
                              void* d_out, int out_size, void* d_ws, size_t ws_size,
                              hipStream_t stream) {
    const float* x      = (const float*)d_in[0];
    const int*   ei     = (const int*)d_in[1];   // [2, E]
    // d_in[2] = edge_len (unused by reference)
    const float* w_qkv  = (const float*)d_in[3];
    const float* w_out  = (const float*)d_in[4];
    const float* b_out  = (const float*)d_in[5];
    const float* ln1_g  = (const float*)d_in[6];
    const float* ln1_b  = (const float*)d_in[7];
    const float* ln2_g  = (const float*)d_in[8];
    const float* ln2_b  = (const float*)d_in[9];
    const float* w_ffn1 = (const float*)d_in[10];
    const float* b_ffn1 = (const float*)d_in[11];
    const float* w_ffn2 = (const float*)d_in[12];
    const float* b_ffn2 = (const float*)d_in[13];
    float* out = (float*)d_out;

    float* ws     = (float*)d_ws;
    float* qkv    = ws;                                  // N*192
    float* scores = qkv    + (size_t)N_NODES * 192;      // E*4
    float* Ssum   = scores + (size_t)N_EDGES * N_HEADS;  // N*4
    float* x1     = Ssum   + (size_t)N_NODES * N_HEADS;  // N*64
    float* hbuf   = x1     + (size_t)N_NODES * 64;       // N*256
    float* stats  = hbuf   + (size_t)N_NODES * 256;      // 4 ints + 4 floats
    int*   gmax   = (int*)stats;
    float* gsum   = stats + N_HEADS;

    hipMemsetAsync(Ssum, 0, (size_t)N_NODES * N_HEADS * sizeof(float), stream);
    k_init_stats<<<1, 32, 0, stream>>>(gmax, gsum);

    const int nb64 = (N_NODES + 63) / 64;    // 782
    const int nb32 = (N_NODES + 31) / 32;    // 1563
    const int nbE  = (N_EDGES * N_HEADS + 255) / 256;  // 12500

    const size_t smem_qkv  = (64 * STRIDE_A + 64 * (96 + 16)) * sizeof(float);   // 46 KB
    const size_t smem_ffn1 = (64 * STRIDE_A + 64 * (128 + 16)) * sizeof(float);  // 54 KB
    const size_t smem_attn = (64 * STRIDE_A + 64 * 80) * sizeof(float);          // 38 KB
    const size_t smem_ffn2 = (32 * 260) * sizeof(float);                         // 33 KB

    // 1) xn = LN1(x); qkv = xn @ w_qkv
    k_ln_gemm<<<nb64, 256, smem_qkv, stream>>>(x, ln1_g, ln1_b, w_qkv, nullptr, qkv, 192, 0);
    // 2) scores + global max per head
    k_scores<<<nbE, 256, 0, stream>>>(qkv, ei, scores, gmax);
    // 3) exp(s-max) -> S[dst][h], global denom
    k_expsum<<<nbE, 256, 0, stream>>>(scores, ei, gmax, Ssum, gsum);
    // 4) x1 = x + (v*S/sum) @ w_out + b_out
    k_attn_out<<<nb64, 256, smem_attn, stream>>>(qkv, Ssum, gsum, w_out, b_out, x, x1);
    // 5) h = gelu(LN2(x1) @ w_ffn1 + b_ffn1)
    k_ln_gemm<<<nb64, 256, smem_ffn1, stream>>>(x1, ln2_g, ln2_b, w_ffn1, b_ffn1, hbuf, 256, 1);
    // 6) out = x1 + h @ w_ffn2 + b_ffn2
    k_ffn2<<<nb32, 256, smem_ffn2, stream>>>(hbuf, w_ffn2, b_ffn2, x1, out);
}